// up_sample_76158360093247
// MI455X (gfx1250) — compile-verified
//
#include <hip/hip_runtime.h>

#define BQ   4
#define NP   4096
#define NUP  16384
#define CH   64
#define ODIM 64
#define EPSW 1e-6f

// padded LDS strides (floats) - chosen for conflict-free WMMA operand fetch
#define NF_S 67   // A tile: bank = (3*row + col) % 64 -> distinct across half-waves
#define W_S  72   // B tile: 2-row offset = 144 % 64 = 16 banks -> disjoint halves

typedef __attribute__((ext_vector_type(2))) float v2f;
typedef __attribute__((ext_vector_type(8))) float v8f;

// ---------------------------------------------------------------------------
// Kernel A: brute-force 3-NN per query (one query per lane), inverse-distance
// weights normalized, idx+weights -> workspace.
// ---------------------------------------------------------------------------
__global__ __launch_bounds__(256) void knn_top3_kernel(
    const float* __restrict__ pos,      // (B, NP, 3)
    const float* __restrict__ pos_up,   // (B, NUP, 3)
    int*  __restrict__ idx_ws,          // (B, NUP, 3)
    float* __restrict__ w_ws)           // (B, NUP, 3)
{
    __shared__ float sx[NP];
    __shared__ float sy[NP];
    __shared__ float sz[NP];
    __shared__ float sp2[NP];

    const int b   = blockIdx.y;
    const int tid = threadIdx.x;
    const float* posb = pos + (size_t)b * NP * 3;

    // Stage pos[b] into LDS as SoA (coalesced global reads, scatter to SoA)
    for (int e = tid; e < NP * 3; e += 256) {
        float v = posb[e];
        int j = e / 3;
        int k = e - 3 * j;
        (k == 0 ? sx : (k == 1 ? sy : sz))[j] = v;
    }
    __syncthreads();
    for (int j = tid; j < NP; j += 256)
        sp2[j] = sx[j] * sx[j] + sy[j] * sy[j] + sz[j] * sz[j];
    __syncthreads();

    const int q = blockIdx.x * 256 + tid;        // one query per lane
    const float* qp = pos_up + ((size_t)b * NUP + q) * 3;
    const float qx = qp[0], qy = qp[1], qz = qp[2];
    const float q2 = qx * qx + qy * qy + qz * qz;

    float b0 = 3.4e38f, b1 = 3.4e38f, b2 = 3.4e38f;
    int   i0 = 0, i1 = 0, i2 = 0;

    // All 32 lanes broadcast-read the same LDS word -> no bank conflicts.
#pragma unroll 4
    for (int j = 0; j < NP; ++j) {
        float dot = qx * sx[j] + qy * sy[j] + qz * sz[j];
        float d   = q2 + sp2[j] - 2.0f * dot;
        if (d < b2) {
            if (d < b1) {
                if (d < b0) { b2 = b1; i2 = i1; b1 = b0; i1 = i0; b0 = d; i0 = j; }
                else        { b2 = b1; i2 = i1; b1 = d;  i1 = j; }
            } else          { b2 = d;  i2 = j; }
        }
    }

    const float w0 = 1.0f / (b0 + EPSW);
    const float w1 = 1.0f / (b1 + EPSW);
    const float w2 = 1.0f / (b2 + EPSW);
    const float inv = 1.0f / (w0 + w1 + w2);

    const size_t base = ((size_t)b * NUP + q) * 3;
    idx_ws[base + 0] = i0; idx_ws[base + 1] = i1; idx_ws[base + 2] = i2;
    w_ws[base + 0] = w0 * inv; w_ws[base + 1] = w1 * inv; w_ws[base + 2] = w2 * inv;
}

// ---------------------------------------------------------------------------
// Kernel B: fused gather-interpolate (128x64 tile -> LDS) + f32 WMMA GEMM
// (tile @ W 64x64) + bias + ReLU.
// ---------------------------------------------------------------------------
__global__ __launch_bounds__(256) void interp_gemm_kernel(
    const float* __restrict__ feature,  // (B, NP, CH)
    const float* __restrict__ Wm,       // (CH, ODIM)
    const float* __restrict__ bias,     // (ODIM)
    const int*  __restrict__ idx_ws,
    const float* __restrict__ w_ws,
    float* __restrict__ out)            // (B, NUP, ODIM)
{
    __shared__ float snf[128 * NF_S];   // interpolated feature tile (A operand)
    __shared__ float sw[CH * W_S];      // weight matrix (B operand)

    const int b   = blockIdx.y;
    const int q0  = blockIdx.x * 128;
    const int tid = threadIdx.x;
    const float* fb = feature + (size_t)b * NP * CH;

    // Stage W (64x64) with padded stride
    for (int e = tid; e < CH * ODIM; e += 256) {
        int r = e >> 6, c = e & 63;
        sw[r * W_S + c] = Wm[e];
    }

    // Build interpolated tile: 128 queries x 64 channels.
    // Consecutive threads -> consecutive channels -> coalesced row gathers
    // (feature is L2-resident: 4 MB << 192 MB).
    for (int e = tid; e < 128 * CH; e += 256) {
        int ql = e >> 6, c = e & 63;
        size_t base = ((size_t)b * NUP + q0 + ql) * 3;
        int   j0 = idx_ws[base + 0], j1 = idx_ws[base + 1], j2 = idx_ws[base + 2];
        float w0 = w_ws[base + 0],  w1 = w_ws[base + 1],  w2 = w_ws[base + 2];
        float v = w0 * fb[(size_t)j0 * CH + c]
                + w1 * fb[(size_t)j1 * CH + c]
                + w2 * fb[(size_t)j2 * CH + c];
        snf[ql * NF_S + c] = v;
    }
    __syncthreads();

    const int wave = tid >> 5;
    const int lane = tid & 31;
    const int lrow = lane & 15;     // M (A) / N (B,C,D) index within half-wave
    const int lhi  = lane >> 4;     // half-wave select

    // 8x4 = 32 output tiles of 16x16; each of the 8 waves computes 4 tiles.
#pragma unroll
    for (int t = 0; t < 4; ++t) {
        const int tile = wave + t * 8;
        const int m0 = (tile & 7) * 16;
        const int n0 = (tile >> 3) * 16;
        v8f acc = {};

        // K = 64 in steps of 4 -> 16 V_WMMA_F32_16X16X4_F32 per tile.
        // A layout (ISA 7.12.2): lane(l) VGPR j holds A[l%16][2*(l/16)+j]
        // B layout mirrors C/D row split: VGPR j holds B[2*(l/16)+j][l%16]
#pragma unroll
        for (int k = 0; k < CH; k += 4) {
            const int ka = k + 2 * lhi;
            v2f a, bm;
            a.x  = snf[(m0 + lrow) * NF_S + ka];
            a.y  = snf[(m0 + lrow) * NF_S + ka + 1];
            bm.x = sw[ka * W_S + n0 + lrow];
            bm.y = sw[(ka + 1) * W_S + n0 + lrow];
            acc = __builtin_amdgcn_wmma_f32_16x16x4_f32(
                /*neg_a=*/false, a, /*neg_b=*/false, bm,
                /*c_mod=*/(short)0, acc, /*reuse_a=*/false, /*reuse_b=*/false);
        }

        // Epilogue: bias + ReLU, coalesced 16-float row segments per half-wave.
        const int col = n0 + lrow;
        const float bb = bias[col];
#pragma unroll
        for (int r = 0; r < 8; ++r) {
            const int row = m0 + r + 8 * lhi;            // C/D VGPR layout
            float v = acc[r] + bb;
            v = v > 0.0f ? v : 0.0f;
            out[((size_t)b * NUP + q0 + row) * ODIM + col] = v;
        }
    }
}

// ---------------------------------------------------------------------------
extern "C" void kernel_launch(void* const* d_in, const int* in_sizes, int n_in,
                              void* d_out, int out_size, void* d_ws, size_t ws_size,
                              hipStream_t stream)
{
    const float* feature = (const float*)d_in[0];   // (4, 4096, 64)
    const float* pos     = (const float*)d_in[1];   // (4, 4096, 3)
    const float* pos_up  = (const float*)d_in[2];   // (4, 16384, 3)
    const float* Wm      = (const float*)d_in[3];   // (64, 64)
    const float* bias    = (const float*)d_in[4];   // (64)
    float* out = (float*)d_out;

    int*   idx_ws = (int*)d_ws;
    float* w_ws   = (float*)((char*)d_ws + (size_t)BQ * NUP * 3 * sizeof(int));

    dim3 gA(NUP / 256, BQ);
    knn_top3_kernel<<<gA, 256, 0, stream>>>(pos, pos_up, idx_ws, w_ws);

    dim3 gB(NUP / 128, BQ);
    interp_gemm_kernel<<<gB, 256, 0, stream>>>(feature, Wm, bias, idx_ws, w_ws, out);
}